// UncertaintyCalibrationLoss_51866025067153
// MI455X (gfx1250) — compile-verified
//
#include <hip/hip_runtime.h>

typedef __attribute__((ext_vector_type(2))) float v2f;
typedef __attribute__((ext_vector_type(8))) float v8f;

#define BDIM 256
#define ROW_V 32000          // vocab per row
#define NV4   (ROW_V / 4)    // 8000 float4 per row
#define NCHUNK (ROW_V / 64)  // 500 wave-chunks of 64 floats

// One workgroup per (b,s) row. Pass 1: row max (float4 strided loads).
// Pass 2: sum exp(x - max) accumulated on the matrix pipe via
// V_WMMA_F32_16X16X4_F32 with B = ones. conf[row] = 1 / sum.
__global__ __launch_bounds__(BDIM) void row_conf_kernel(
    const float* __restrict__ logits, float* __restrict__ conf) {
  const int row  = blockIdx.x;                 // 0..1023
  const int tid  = threadIdx.x;
  const int lane = tid & 31;
  const int wid  = tid >> 5;                   // 8 waves / block
  const float* rowp = logits + (long long)row * ROW_V;

  __shared__ float red[BDIM];

  // ---- pass 1: row max ----
  float m = -3.402823466e38f;
  const float4* r4 = (const float4*)rowp;
  for (int i = tid; i < NV4; i += BDIM) {
    float4 v = r4[i];
    m = fmaxf(m, fmaxf(fmaxf(v.x, v.y), fmaxf(v.z, v.w)));
  }
  red[tid] = m;
  __syncthreads();
  for (int s = BDIM / 2; s > 0; s >>= 1) {
    if (tid < s) red[tid] = fmaxf(red[tid], red[tid + s]);
    __syncthreads();
  }
  const float rowmax = red[0];
  __syncthreads();

  // ---- pass 2: sum of exp(x - rowmax), WMMA-accumulated ----
  // Each wave: 64 consecutive floats per chunk (2 per lane = 16x4 f32 A tile).
  v8f acc = {0.f, 0.f, 0.f, 0.f, 0.f, 0.f, 0.f, 0.f};
  const v2f ones = {1.0f, 1.0f};
  for (int chunk = wid; chunk < NCHUNK; chunk += BDIM / 32) {
    const float2 x = *(const float2*)(rowp + chunk * 64 + lane * 2);
    v2f a;
    a.x = __expf(x.x - rowmax);
    a.y = __expf(x.y - rowmax);
    // D[m,n] = sum_k A[m,k] * 1 + C[m,n]  (accumulates 4 exp values per row m)
    acc = __builtin_amdgcn_wmma_f32_16x16x4_f32(
        /*neg_a=*/false, a, /*neg_b=*/false, ones,
        /*c_mod=*/(short)0, acc, /*reuse_a=*/false, /*reuse_b=*/false);
  }
  // Rows m=0..7 live in acc[0..7] on lanes 0-15, rows 8..15 on lanes 16-31.
  float t = acc[0] + acc[1] + acc[2] + acc[3] + acc[4] + acc[5] + acc[6] + acc[7];
  t += __shfl_xor(t, 16, 32);                  // combine the two lane halves
  if (lane == 0) red[wid] = t;
  __syncthreads();
  if (tid == 0) {
    float total = 0.f;
    for (int w = 0; w < BDIM / 32; ++w) total += red[w];
    conf[row] = 1.0f / total;                  // max softmax prob of the row
  }
}

// Single block: per-sample averages, uncertainty normalization,
// keyword penalties, final scalar. Wave w handles sample b = w (B = 8).
__global__ __launch_bounds__(256) void finalize_kernel(
    const float* __restrict__ conf,        // [1024] per-row confidences
    const float* __restrict__ epistemic,   // [8*14]
    const float* __restrict__ aleatoric,   // [8*14]
    const float* __restrict__ consistency, // [8]
    const int* __restrict__ tokens,        // [8*128]
    const int* __restrict__ hi_ids,        // [6]
    const int* __restrict__ lo_ids,        // [6]
    const int* __restrict__ un_ids,        // [5]
    float* __restrict__ out) {
  __shared__ float s_avgc[8], s_tu[8], s_p[8];
  const int tid  = threadIdx.x;
  const int b    = tid >> 5;   // 8 waves -> 8 samples
  const int lane = tid & 31;

  // mean over S=128 of per-row confidence
  float s = 0.f;
  for (int k = 0; k < 4; ++k) s += conf[b * 128 + lane + 32 * k];
  for (int o = 16; o > 0; o >>= 1) s += __shfl_xor(s, o, 32);
  if (lane == 0) s_avgc[b] = s * (1.0f / 128.0f);

  // total uncertainty = mean(epistemic) + mean(aleatoric), C = 14
  if (lane == 1) {
    float se = 0.f, sa = 0.f;
    for (int j = 0; j < 14; ++j) {
      se += epistemic[b * 14 + j];
      sa += aleatoric[b * 14 + j];
    }
    s_tu[b] = se * (1.0f / 14.0f) + sa * (1.0f / 14.0f);
  }
  __syncthreads();

  // keyword counts: each lane checks 4 tokens against the 3 id sets
  int ch = 0, cl = 0, cu = 0;
  for (int k = 0; k < 4; ++k) {
    const int tok = tokens[b * 128 + lane * 4 + k];
    bool ih = false, il = false, iu = false;
    for (int j = 0; j < 6; ++j) {
      ih = ih || (tok == hi_ids[j]);
      il = il || (tok == lo_ids[j]);
    }
    for (int j = 0; j < 5; ++j) iu = iu || (tok == un_ids[j]);
    ch += ih ? 1 : 0; cl += il ? 1 : 0; cu += iu ? 1 : 0;
  }
  for (int o = 16; o > 0; o >>= 1) {
    ch += __shfl_xor(ch, o, 32);
    cl += __shfl_xor(cl, o, 32);
    cu += __shfl_xor(cu, o, 32);
  }
  if (lane == 0) {
    const float tu = s_tu[b];
    float p = 0.f;
    if (tu > 0.6f && ch > cl) p += 0.5f;
    if (tu < 0.3f && cl > ch) p += 0.3f;
    if (consistency[b] < 0.5f && cu == 0) p += 0.4f;
    s_p[b] = p;
  }
  __syncthreads();

  if (tid == 0) {
    float maxtu = s_tu[0];
    for (int i = 1; i < 8; ++i) maxtu = fmaxf(maxtu, s_tu[i]);
    const float inv = 1.0f / (maxtu + 1e-8f);
    float calib = 0.f, pen = 0.f;
    for (int i = 0; i < 8; ++i) {
      const float target = 1.0f - s_tu[i] * inv;
      const float d = s_avgc[i] - target;
      calib += d * d;
      pen += s_p[i];
    }
    out[0] = calib * (1.0f / 8.0f) + 0.1f * (pen * (1.0f / 8.0f));
  }
}

extern "C" void kernel_launch(void* const* d_in, const int* in_sizes, int n_in,
                              void* d_out, int out_size, void* d_ws, size_t ws_size,
                              hipStream_t stream) {
  const float* logits      = (const float*)d_in[0];  // [8,128,32000] f32
  const float* epistemic   = (const float*)d_in[1];  // [8,14] f32
  const float* aleatoric   = (const float*)d_in[2];  // [8,14] f32
  const float* consistency = (const float*)d_in[3];  // [8] f32
  const int*   tokens      = (const int*)d_in[4];    // [8,128] i32
  const int*   hi_ids      = (const int*)d_in[5];    // [6] i32
  const int*   lo_ids      = (const int*)d_in[6];    // [6] i32
  const int*   un_ids      = (const int*)d_in[7];    // [5] i32
  float*       out         = (float*)d_out;          // scalar f32
  float*       conf        = (float*)d_ws;           // 1024 floats scratch

  row_conf_kernel<<<8 * 128, BDIM, 0, stream>>>(logits, conf);
  finalize_kernel<<<1, 256, 0, stream>>>(conf, epistemic, aleatoric, consistency,
                                         tokens, hi_ids, lo_ids, un_ids, out);
}